// Wn0110C1_38268158608162
// MI455X (gfx1250) — compile-verified
//
#include <hip/hip_runtime.h>
#include <math.h>

#define NT 365
#define NS 300
#define NH 128
#define NG 32
#define NR 8
#define ROWS (NT * NS)        // 109500
#define ROWSP 109504          // ROWS padded to multiple of 16
#define K1PAD 40              // 38 inputs padded to multiple of 4
#define K3PAD 36              // 34 inputs padded to multiple of 4
#define HSTR 264              // 256 hidden + 8 pad (bank-conflict relief)

typedef __attribute__((ext_vector_type(2))) float v2f;
typedef __attribute__((ext_vector_type(8))) float v8f;

// D = A(16x4 f32) * B(4x16 f32) + C(16x16 f32), exact f32 per reference.
__device__ __forceinline__ v8f wmma_k4(v2f a, v2f b, v8f c) {
  return __builtin_amdgcn_wmma_f32_16x16x4_f32(
      /*neg_a=*/false, a, /*neg_b=*/false, b,
      /*c_mod=*/(short)0, c, /*reuse_a=*/false, /*reuse_b=*/false);
}

// Branch-free transcendentals (v_exp_f32 / v_rcp_f32; no exec-mask ladders).
__device__ __forceinline__ float exp_fast(float x) { return __expf(x); }
__device__ __forceinline__ float rcp_fast(float x) { return __builtin_amdgcn_rcpf(x); }
__device__ __forceinline__ float tanh_fast(float x) {
  // tanh(x) = 1 - 2/(e^{2x}+1); e->inf => 1, e->0 => -1 (correct limits)
  const float e = __expf(2.f * x);
  return 1.f - 2.f * rcp_fast(e + 1.f);
}
__device__ __forceinline__ float sigmoid_fast(float x) {
  return rcp_fast(1.f + __expf(-x));
}

// ---------------------------------------------------------------------------
// Kernel 0: zero-pad weight matrices so WMMA B-fragment loads are branch-free
// 64-bit vector loads. W1 [256,38]->[256,40]; cW [384,34]->[384,36].
// ---------------------------------------------------------------------------
__global__ __launch_bounds__(256)
void pad_weights_kernel(const float* __restrict__ tW1, const float* __restrict__ cW,
                        float* __restrict__ W1p, float* __restrict__ cWp) {
  const int idx = blockIdx.x * 256 + threadIdx.x;
  if (idx < 256 * K1PAD) {
    const int i = idx / K1PAD, k = idx % K1PAD;
    W1p[idx] = (k < 38) ? tW1[i * 38 + k] : 0.f;
  }
  const int idx2 = idx - 256 * K1PAD;
  if (idx2 >= 0 && idx2 < 384 * K3PAD) {
    const int i = idx2 / K3PAD, k = idx2 % K3PAD;
    cWp[idx2] = (k < 34) ? cW[i * 34 + k] : 0.f;
  }
}

// ---------------------------------------------------------------------------
// Kernel 1: per-site static MLPs -> gates kp,ks,kg,gp,gL,qb,ga(softmax), r
// One block per site, 128 threads (= NH buckets). Tiny GEMMs, plain VALU.
// ---------------------------------------------------------------------------
__global__ __launch_bounds__(128)
void static_gates_kernel(const float* __restrict__ xc,
                         const float* __restrict__ R_w1, const float* __restrict__ R_b1,
                         const float* __restrict__ R_w2, const float* __restrict__ R_b2,
                         const float* __restrict__ W_w1, const float* __restrict__ W_b1,
                         const float* __restrict__ W_w2, const float* __restrict__ W_b2,
                         float* __restrict__ kp, float* __restrict__ ks,
                         float* __restrict__ kg, float* __restrict__ gp,
                         float* __restrict__ gL, float* __restrict__ qb,
                         float* __restrict__ ga, float* __restrict__ r) {
  const int s = blockIdx.x, tid = threadIdx.x;
  __shared__ float s_xc[NG];
  __shared__ float s_h[256];
  __shared__ float s_tmp[128];

  if (tid < NG) s_xc[tid] = xc[s * NG + tid];
  __syncthreads();

  // --- W branch: hidden = tanh(xc @ W1^T + b1) ---
  for (int i = tid; i < 256; i += 128) {
    float acc = W_b1[i];
    const float* wr = W_w1 + i * NG;
    for (int k = 0; k < NG; ++k) acc += s_xc[k] * wr[k];
    s_h[i] = tanh_fast(acc);
  }
  __syncthreads();

  float raw_ga = 0.f;
  const int gi = s * NH + tid;
  for (int g = 0; g < 7; ++g) {
    const int o = g * 128 + tid;
    float acc = W_b2[o];
    const float* wr = W_w2 + (size_t)o * 256;
    for (int k = 0; k < 256; ++k) acc += s_h[k] * wr[k];
    switch (g) {
      case 0: kp[gi] = sigmoid_fast(acc); break;
      case 1: ks[gi] = sigmoid_fast(acc); break;
      case 2: kg[gi] = sigmoid_fast(acc); break;
      case 3: gp[gi] = sigmoid_fast(acc); break;
      case 4: gL[gi] = exp_fast(acc) * 2.f; break;
      case 5: qb[gi] = fmaxf(acc, 0.f); break;
      case 6: raw_ga = acc; break;
    }
  }
  // softmax over the 128 buckets
  s_tmp[tid] = raw_ga;
  __syncthreads();
  for (int off = 64; off > 0; off >>= 1) {
    if (tid < off) s_tmp[tid] = fmaxf(s_tmp[tid], s_tmp[tid + off]);
    __syncthreads();
  }
  const float mx = s_tmp[0];
  __syncthreads();
  const float e = exp_fast(raw_ga - mx);
  s_tmp[tid] = e;
  __syncthreads();
  for (int off = 64; off > 0; off >>= 1) {
    if (tid < off) s_tmp[tid] += s_tmp[tid + off];
    __syncthreads();
  }
  ga[gi] = e / s_tmp[0];
  __syncthreads();

  // --- R branch: routing weights r = relu(MLP(xc)) reshaped [NH][NR] ---
  for (int i = tid; i < 256; i += 128) {
    float acc = R_b1[i];
    const float* wr = R_w1 + i * NG;
    for (int k = 0; k < NG; ++k) acc += s_xc[k] * wr[k];
    s_h[i] = tanh_fast(acc);
  }
  __syncthreads();
  for (int j = 0; j < NR; ++j) {
    const int o = tid * NR + j;   // h*nr + i per reference reshape
    float acc = R_b2[o];
    const float* wr = R_w2 + (size_t)o * 256;
    for (int k = 0; k < 256; ++k) acc += s_h[k] * wr[k];
    r[(size_t)s * NH * NR + o] = fmaxf(acc, 0.f);
  }
}

// ---------------------------------------------------------------------------
// Kernel 2: WMMA time-varying MLP. One wave per 16-row strip of (t,s) rows.
// Stage 1: H[16,256] = tanh(XT[16,40p] @ W1p^T + b1)    (f32 WMMA, K=4)
// Stage 2: v[16,384] = H @ W2^T + b2 -> vi/ve/vm gates
// Stage 3: wC[16,384] = XTc[16,36p] @ cWp^T + cb -> cp/cs/cg = exp(.)
// All fragment loads are unconditional b64s; stores unguarded (padded arrays);
// activations are branch-free v_exp/v_rcp sequences.
// ---------------------------------------------------------------------------
__global__ __launch_bounds__(32)
void time_mlp_wmma_kernel(const float* __restrict__ x, const float* __restrict__ xc,
                          const float* __restrict__ W1p, const float* __restrict__ tB1,
                          const float* __restrict__ tW2, const float* __restrict__ tB2,
                          const float* __restrict__ cWp, const float* __restrict__ cB,
                          float* __restrict__ vi, float* __restrict__ ve,
                          float* __restrict__ vm, float* __restrict__ cp,
                          float* __restrict__ cs, float* __restrict__ cg) {
  __shared__ float s_in[16 * K1PAD];   // input strip (padded K)
  __shared__ float s_h[16 * HSTR];     // hidden strip

  const int lane = threadIdx.x;
  const int half = lane >> 4;          // 0: K offset 0/1, 1: K offset 2/3
  const int l16 = lane & 15;
  const int r0 = blockIdx.x * 16;

  // ---- load XT = [x(6), xc(32), pad(2)] for 16 rows ----
  for (int idx = lane; idx < 16 * K1PAD; idx += 32) {
    const int m = idx / K1PAD, k = idx % K1PAD;
    int row = r0 + m; if (row >= ROWS) row = ROWS - 1;
    const int s = row % NS;
    float v = 0.f;
    if (k < 6) v = x[(size_t)row * 6 + k];
    else if (k < 6 + NG) v = xc[s * NG + (k - 6)];
    s_in[idx] = v;
  }
  __syncthreads();

  // ---- stage 1: hidden 256, 16 N-tiles x 10 K-steps ----
  for (int nt = 0; nt < 16; ++nt) {
    const int n = nt * 16 + l16;
    const float bias = tB1[n];
    v8f acc; for (int j = 0; j < 8; ++j) acc[j] = bias;
#pragma unroll
    for (int k0 = 0; k0 < K1PAD; k0 += 4) {
      const int ka = k0 + 2 * half;                       // even -> 8B aligned
      const v2f a = *(const v2f*)&s_in[l16 * K1PAD + ka]; // ds_load_b64
      const v2f b = *(const v2f*)&W1p[n * K1PAD + ka];    // global_load_b64
      acc = wmma_k4(a, b, acc);
    }
#pragma unroll
    for (int j = 0; j < 8; ++j) {
      const int m = j + 8 * half;
      s_h[m * HSTR + n] = tanh_fast(acc[j]);
    }
  }
  __syncthreads();

  // ---- stage 2: v = H @ W2^T + b2 (384 out), fused gate activations ----
  for (int nt = 0; nt < 24; ++nt) {
    const int n = nt * 16 + l16;
    const float bias = tB2[n];
    v8f acc; for (int j = 0; j < 8; ++j) acc[j] = bias;
#pragma unroll 8
    for (int k0 = 0; k0 < 256; k0 += 4) {
      const int ka = k0 + 2 * half;
      const v2f a = *(const v2f*)&s_h[l16 * HSTR + ka];
      const v2f b = *(const v2f*)&tW2[(size_t)n * 256 + ka];
      acc = wmma_k4(a, b, acc);
    }
    const int g = n >> 7, nh = n & 127;
    float* dst = (g == 0) ? vi : ((g == 1) ? ve : vm);
#pragma unroll
    for (int j = 0; j < 8; ++j) {
      const int row = r0 + j + 8 * half;                  // < ROWSP, unguarded
      float val = acc[j];
      if (g == 0)      val = fminf(fmaxf(val * (1.f / 3.f) + 0.5f, 0.f), 1.f); // hardsigmoid(2v)
      else if (g == 1) val = fmaxf(val, 0.f) * 2.f;                            // relu*2
      else             val = exp_fast(val);                                    // exp
      dst[(size_t)row * NH + nh] = val;
    }
  }
  __syncthreads();

  // ---- stage 3: wC linear layer; rebuild s_in as XTc = [T1,T2,xc,pad2] ----
  for (int idx = lane; idx < 16 * K3PAD; idx += 32) {
    const int m = idx / K3PAD, k = idx % K3PAD;
    int row = r0 + m; if (row >= ROWS) row = ROWS - 1;
    const int s = row % NS;
    float v = 0.f;
    if (k == 0) v = x[(size_t)row * 6 + 2];
    else if (k == 1) v = x[(size_t)row * 6 + 3];
    else if (k < 34) v = xc[s * NG + (k - 2)];
    s_in[m * K3PAD + k] = v;
  }
  __syncthreads();
  for (int nt = 0; nt < 24; ++nt) {
    const int n = nt * 16 + l16;
    const float bias = cB[n];
    v8f acc; for (int j = 0; j < 8; ++j) acc[j] = bias;
#pragma unroll
    for (int k0 = 0; k0 < K3PAD; k0 += 4) {
      const int ka = k0 + 2 * half;
      const v2f a = *(const v2f*)&s_in[l16 * K3PAD + ka];
      const v2f b = *(const v2f*)&cWp[n * K3PAD + ka];
      acc = wmma_k4(a, b, acc);
    }
    const int g = n >> 7, nh = n & 127;
    float* dst = (g == 0) ? cp : ((g == 1) ? cs : cg);
#pragma unroll
    for (int j = 0; j < 8; ++j) {
      const int row = r0 + j + 8 * half;
      dst[(size_t)row * NH + nh] = exp_fast(acc[j]);
    }
  }
}

// ---------------------------------------------------------------------------
// Kernel 3: sequential bucket-state scan over t. One block per site,
// thread = bucket. Emits Q1T,Q2T,Q3T.
// ---------------------------------------------------------------------------
__global__ __launch_bounds__(128)
void scan_kernel(const float* __restrict__ x,
                 const float* __restrict__ vi, const float* __restrict__ ve,
                 const float* __restrict__ vm,
                 const float* __restrict__ kp_, const float* __restrict__ ks_,
                 const float* __restrict__ kg_, const float* __restrict__ gp_,
                 const float* __restrict__ gL_, const float* __restrict__ qb_,
                 float* __restrict__ Q1, float* __restrict__ Q2, float* __restrict__ Q3) {
  const int s = blockIdx.x, h = threadIdx.x;
  const int gi = s * NH + h;
  const float kp = kp_[gi], ks = ks_[gi], kg = kg_[gi];
  const float gp = gp_[gi], gL = gL_[gi], qb = qb_[gi];
  float Sf = 0.f, Ss = 0.f, Sg = 0.f;
  for (int t = 0; t < NT; ++t) {
    const size_t row = (size_t)t * NS + s;
    const float P = x[row * 6 + 0], E = x[row * 6 + 1];
    const float T1 = x[row * 6 + 2], T2 = x[row * 6 + 3];
    const float denom = T2 - T1;
    float ratio = (T1 + T2) / ((denom == 0.f) ? 1.f : denom);
    ratio = fminf(fmaxf(ratio, -1.f), 1.f);
    float vf = acosf(ratio) * (1.f / 3.1415f);
    vf = (T1 >= 0.f) ? 0.f : ((T2 <= 0.f) ? 1.f : vf);
    const float Psk = P * vf;
    const size_t ri = row * NH + h;
    const float Plk = P * (1.f - vf) * vi[ri];
    const float Evk = E * ve[ri];
    const float vmk = vm[ri];
    const float qf = fminf(Sf + Psk, vmk);
    Sf = fmaxf(Sf + Psk - vmk, 0.f);
    const float H = fmaxf(Ss + Plk + qf - Evk, 0.f);
    const float qp = fmaxf(kp * (H - gL), 0.f);
    const float qs = ks * fminf(H, gL);
    Ss = H - qp - qs;
    const float qso = qs * (1.f - gp);
    const float qsg = qs * gp;
    const float qg = kg * (Sg + qsg) + qb;
    Sg = (1.f - kg) * (Sg + qsg) - qb;
    Q1[ri] = qp; Q2[ri] = qso; Q3[ri] = qg;
  }
}

// ---------------------------------------------------------------------------
// Kernel 4: causal 8-tap routing conv + ga-weighted reduction -> outQ, outC.
// One block per (t,s), thread = bucket, LDS tree reduction over 128.
// ---------------------------------------------------------------------------
__global__ __launch_bounds__(128)
void route_kernel(const float* __restrict__ Q1, const float* __restrict__ Q2,
                  const float* __restrict__ Q3,
                  const float* __restrict__ cp, const float* __restrict__ cs,
                  const float* __restrict__ cg,
                  const float* __restrict__ r, const float* __restrict__ ga,
                  float* __restrict__ out) {
  const int bid = blockIdx.x;
  const int t = bid / NS, s = bid % NS;
  const int h = threadIdx.x;
  const float* rh = r + (size_t)s * NH * NR + h * NR;
  float accQ = 0.f, accC = 0.f;
  for (int i = 0; i < NR; ++i) {
    const int ti = t - i;
    if (ti < 0) break;   // causal zero padding
    const size_t ri = ((size_t)ti * NS + s) * NH + h;
    const float w = rh[i];
    const float q1 = Q1[ri], q2 = Q2[ri], q3 = Q3[ri];
    accQ += (q1 + q2 + q3) * w;
    accC += (q1 * cp[ri] * 0.1f + q2 * cs[ri] + q3 * cg[ri]) * w;
  }
  const float g = ga[s * NH + h];
  accQ *= g; accC *= g;
  __shared__ float rQ[128], rC[128];
  rQ[h] = accQ; rC[h] = accC;
  __syncthreads();
  for (int off = 64; off > 0; off >>= 1) {
    if (h < off) { rQ[h] += rQ[h + off]; rC[h] += rC[h + off]; }
    __syncthreads();
  }
  if (h == 0) {
    out[bid] = rQ[0];
    out[ROWS + bid] = rC[0] / rQ[0];
  }
}

// ---------------------------------------------------------------------------
extern "C" void kernel_launch(void* const* d_in, const int* in_sizes, int n_in,
                              void* d_out, int out_size, void* d_ws, size_t ws_size,
                              hipStream_t stream) {
  const float* x      = (const float*)d_in[0];
  const float* xc     = (const float*)d_in[1];
  const float* fcR_w1 = (const float*)d_in[2];
  const float* fcR_b1 = (const float*)d_in[3];
  const float* fcR_w2 = (const float*)d_in[4];
  const float* fcR_b2 = (const float*)d_in[5];
  const float* fcW_w1 = (const float*)d_in[6];
  const float* fcW_b1 = (const float*)d_in[7];
  const float* fcW_w2 = (const float*)d_in[8];
  const float* fcW_b2 = (const float*)d_in[9];
  const float* fcT_w1 = (const float*)d_in[10];
  const float* fcT_b1 = (const float*)d_in[11];
  const float* fcT_w2 = (const float*)d_in[12];
  const float* fcT_b2 = (const float*)d_in[13];
  const float* fcCT_w = (const float*)d_in[14];
  const float* fcCT_b = (const float*)d_in[15];
  float* out = (float*)d_out;

  // workspace layout (floats)
  float* ws = (float*)d_ws;
  const size_t RHP = (size_t)ROWSP * NH;  // padded gate arrays
  const size_t RH  = (size_t)ROWS * NH;
  float* vi = ws + 0 * RHP;
  float* ve = ws + 1 * RHP;
  float* vm = ws + 2 * RHP;
  float* cp = ws + 3 * RHP;
  float* cs = ws + 4 * RHP;
  float* cg = ws + 5 * RHP;
  float* Q1 = ws + 6 * RHP;
  float* Q2 = ws + 6 * RHP + 1 * RH;
  float* Q3 = ws + 6 * RHP + 2 * RH;
  float* gates = ws + 6 * RHP + 3 * RH;
  const size_t SH = (size_t)NS * NH;      // 38,400
  float* kp  = gates + 0 * SH;
  float* ks  = gates + 1 * SH;
  float* kg  = gates + 2 * SH;
  float* gp  = gates + 3 * SH;
  float* gL  = gates + 4 * SH;
  float* qb  = gates + 5 * SH;
  float* ga  = gates + 6 * SH;
  float* rr  = gates + 7 * SH;            // [NS][NH][NR]
  float* W1p = rr + (size_t)NS * NH * NR; // [256][40]
  float* cWp = W1p + 256 * K1PAD;         // [384][36]

  const int pad_elems = 256 * K1PAD + 384 * K3PAD;
  pad_weights_kernel<<<(pad_elems + 255) / 256, 256, 0, stream>>>(
      fcT_w1, fcCT_w, W1p, cWp);

  static_gates_kernel<<<NS, 128, 0, stream>>>(
      xc, fcR_w1, fcR_b1, fcR_w2, fcR_b2, fcW_w1, fcW_b1, fcW_w2, fcW_b2,
      kp, ks, kg, gp, gL, qb, ga, rr);

  const int strips = ROWSP / 16;          // 6844 waves
  time_mlp_wmma_kernel<<<strips, 32, 0, stream>>>(
      x, xc, W1p, fcT_b1, fcT_w2, fcT_b2, cWp, fcCT_b,
      vi, ve, vm, cp, cs, cg);

  scan_kernel<<<NS, 128, 0, stream>>>(
      x, vi, ve, vm, kp, ks, kg, gp, gL, qb, Q1, Q2, Q3);

  route_kernel<<<ROWS, 128, 0, stream>>>(
      Q1, Q2, Q3, cp, cs, cg, rr, ga, out);
}